// LIFNode_5196910428185
// MI455X (gfx1250) — compile-verified
//
#include <hip/hip_runtime.h>

// LIF multi-step forward, T=4, hard reset, decay_input, forward==Heaviside.
// Memory-bound streaming kernel: 1 GiB traffic -> ~46 us floor @ 23.3 TB/s.
// Strategy: 128-bit vectorized, non-temporal (streaming) loads/stores,
// T fully unrolled so all 4 plane loads issue back-to-back per thread.

typedef __attribute__((ext_vector_type(4))) float v4f;

#define TSTEPS 4

__global__ __launch_bounds__(256) void lif_fwd_vec4(
    const float* __restrict__ x, float* __restrict__ out, long long plane4) {
  long long i = (long long)blockIdx.x * blockDim.x + threadIdx.x;
  if (i >= plane4) return;

  const v4f* __restrict__ xv = (const v4f*)x;
  v4f* __restrict__ ov = (v4f*)out;

  // Issue all 4 timestep loads up front (independent B128 NT loads).
  v4f xt[TSTEPS];
#pragma unroll
  for (int t = 0; t < TSTEPS; ++t)
    xt[t] = __builtin_nontemporal_load(xv + (i + (long long)t * plane4));

  v4f v = {0.0f, 0.0f, 0.0f, 0.0f};  // V_RESET = 0
#pragma unroll
  for (int t = 0; t < TSTEPS; ++t) {
    v4f vc = (v + xt[t]) * 0.5f;  // v + (x - v)/tau, tau=2, reset=0
    v4f s;
#pragma unroll
    for (int c = 0; c < 4; ++c) {
      bool fire = vc[c] >= 1.0f;  // v_th = 1
      s[c] = fire ? 1.0f : 0.0f;  // forward spike == heaviside
      v[c] = fire ? 0.0f : vc[c]; // hard reset
    }
    __builtin_nontemporal_store(s, ov + (i + (long long)t * plane4));
  }
}

// Scalar tail for plane sizes not divisible by 4 (not hit for this shape).
__global__ __launch_bounds__(256) void lif_fwd_tail(
    const float* __restrict__ x, float* __restrict__ out,
    long long start, long long plane) {
  long long i = start + (long long)blockIdx.x * blockDim.x + threadIdx.x;
  if (i >= plane) return;
  float v = 0.0f;
#pragma unroll
  for (int t = 0; t < TSTEPS; ++t) {
    float xt = __builtin_nontemporal_load(x + (i + (long long)t * plane));
    float vc = (v + xt) * 0.5f;
    bool fire = vc >= 1.0f;
    __builtin_nontemporal_store(fire ? 1.0f : 0.0f,
                                out + (i + (long long)t * plane));
    v = fire ? 0.0f : vc;
  }
}

extern "C" void kernel_launch(void* const* d_in, const int* in_sizes, int n_in,
                              void* d_out, int out_size, void* d_ws, size_t ws_size,
                              hipStream_t stream) {
  const float* x = (const float*)d_in[0];
  float* out = (float*)d_out;

  long long total = (long long)in_sizes[0];     // T * B * N
  long long plane = total / TSTEPS;             // B * N (elements per timestep)
  long long plane4 = plane / 4;                 // float4 columns

  if (plane4 > 0) {
    long long blocks = (plane4 + 255) / 256;
    lif_fwd_vec4<<<dim3((unsigned)blocks), dim3(256), 0, stream>>>(x, out, plane4);
  }
  long long covered = plane4 * 4;
  long long rem = plane - covered;
  if (rem > 0) {
    long long blocks = (rem + 255) / 256;
    lif_fwd_tail<<<dim3((unsigned)blocks), dim3(256), 0, stream>>>(x, out, covered, plane);
  }
}